// DynamicScaleAttention_87625922773358
// MI455X (gfx1250) — compile-verified
//
#include <hip/hip_runtime.h>

typedef __attribute__((ext_vector_type(2))) float v2f;
typedef __attribute__((ext_vector_type(8))) float v8f;

#define B_SZ   8
#define N_SEQ  1024
#define D_MOD  768
#define H_CNT  12
#define D_HEAD 64
#define N_QKV  2304

__device__ __forceinline__ v8f wmma4(v2f a, v2f b, v8f c) {
  // D = A(16x4 f32) * B(4x16 f32) + C(16x16 f32)
  return __builtin_amdgcn_wmma_f32_16x16x4_f32(false, a, false, b, (short)0, c,
                                               false, false);
}

// ---------------------------------------------------------------------------
// Kernel 1: QKV projection  X[8192,768] @ Wqkv[768,2304], scatter to
// Q/K/V in [B,H,N,Dh] layout.  WG = 8 waves (4 row-groups x 2 col-groups)
// -> 64x128 tile.  grid = (2304/128, 8192/64)
// ---------------------------------------------------------------------------
__global__ __launch_bounds__(256) void qkv_gemm_kernel(
    const float* __restrict__ X, const float* __restrict__ W,
    float* __restrict__ Qo, float* __restrict__ Ko, float* __restrict__ Vo)
{
  const int lane = threadIdx.x & 31;
  const int wave = threadIdx.x >> 5;
  const int l    = lane & 15;
  const int hf   = lane >> 4;
  const int row0 = blockIdx.y * 64 + (wave >> 1) * 16;
  const int col0 = blockIdx.x * 128 + (wave & 1) * 64;

  v8f acc[4] = {{}, {}, {}, {}};
  const float* arow = X + (size_t)(row0 + l) * D_MOD;
  for (int k = 0; k < D_MOD; k += 4) {
    const int kk = k + 2 * hf;
    const v2f a = *(const v2f*)(arow + kk);
    const float* b0 = W + (size_t)kk * N_QKV + col0 + l;
    const float* b1 = b0 + N_QKV;
    #pragma unroll
    for (int t = 0; t < 4; ++t) {
      v2f b; b.x = b0[16 * t]; b.y = b1[16 * t];
      acc[t] = wmma4(a, b, acc[t]);
    }
  }
  #pragma unroll
  for (int t = 0; t < 4; ++t) {
    const int cbase = col0 + 16 * t;         // global output column base
    const int t3  = cbase / D_MOD;           // 0=Q 1=K 2=V
    const int rem = cbase - t3 * D_MOD;
    const int h   = rem >> 6;
    const int dh  = (rem & 63) + l;
    float* dst = (t3 == 0) ? Qo : ((t3 == 1) ? Ko : Vo);
    #pragma unroll
    for (int j = 0; j < 8; ++j) {
      const int r = row0 + j + 8 * hf;       // global row (b*N + n)
      const int b = r >> 10;
      const int n = r & (N_SEQ - 1);
      dst[((size_t)(b * H_CNT + h) * N_SEQ + n) * D_HEAD + dh] = acc[t][j];
    }
  }
}

__global__ void init_absmax_kernel(unsigned* __restrict__ a) {
  if (threadIdx.x < B_SZ * H_CNT) a[threadIdx.x] = 0u;
}

// ---------------------------------------------------------------------------
// Kernel 2: raw scores S = Q @ K^T per (b,h); write into w-region of d_out,
// accumulate per-(b,h) abs-max.  grid = (1024/128, 1024/64, 96)
// ---------------------------------------------------------------------------
__global__ __launch_bounds__(256) void scores_kernel(
    const float* __restrict__ Q, const float* __restrict__ K,
    float* __restrict__ S, unsigned* __restrict__ amax)
{
  const int bh = blockIdx.z;
  const float* q  = Q + (size_t)bh * N_SEQ * D_HEAD;
  const float* km = K + (size_t)bh * N_SEQ * D_HEAD;
  float* s = S + (size_t)bh * N_SEQ * N_SEQ;

  const int lane = threadIdx.x & 31;
  const int wave = threadIdx.x >> 5;
  const int l    = lane & 15;
  const int hf   = lane >> 4;
  const int row0 = blockIdx.y * 64 + (wave >> 1) * 16;
  const int col0 = blockIdx.x * 128 + (wave & 1) * 64;

  v8f acc[4] = {{}, {}, {}, {}};
  const float* arow = q  + (size_t)(row0 + l) * D_HEAD;
  const float* brow = km + (size_t)(col0 + l) * D_HEAD;
  #pragma unroll 4
  for (int k = 0; k < D_HEAD; k += 4) {
    const int kk = k + 2 * hf;
    const v2f a = *(const v2f*)(arow + kk);
    #pragma unroll
    for (int t = 0; t < 4; ++t) {
      const v2f b = *(const v2f*)(brow + (size_t)(16 * t) * D_HEAD + kk);
      acc[t] = wmma4(a, b, acc[t]);
    }
  }
  float mloc = 0.0f;
  #pragma unroll
  for (int t = 0; t < 4; ++t) {
    #pragma unroll
    for (int j = 0; j < 8; ++j) {
      const float v = acc[t][j];
      mloc = fmaxf(mloc, fabsf(v));
      s[(size_t)(row0 + j + 8 * hf) * N_SEQ + col0 + 16 * t + l] = v;
    }
  }
  #pragma unroll
  for (int off = 16; off > 0; off >>= 1)
    mloc = fmaxf(mloc, __shfl_xor(mloc, off, 32));
  if (lane == 0) atomicMax(&amax[bh], __float_as_uint(mloc)); // vals >= 0
}

// ---------------------------------------------------------------------------
// Kernel 3: in-place row softmax of the scaled scores.  One 256-thread block
// per row (4 elements/thread).  grid = 98304
// ---------------------------------------------------------------------------
__global__ __launch_bounds__(256) void softmax_kernel(
    float* __restrict__ Wm, const unsigned* __restrict__ amax,
    const float* __restrict__ alpha)
{
  __shared__ float red[8];
  const int row = blockIdx.x;
  const int bh  = row >> 10;
  float* wr = Wm + (size_t)row * N_SEQ;
  const float scale = alpha[0] / fmaxf(__uint_as_float(amax[bh]), 1.0f);
  const int tid  = threadIdx.x;
  const int lane = tid & 31, wv = tid >> 5;

  float4 v = ((const float4*)wr)[tid];
  v.x *= scale; v.y *= scale; v.z *= scale; v.w *= scale;
  float m = fmaxf(fmaxf(v.x, v.y), fmaxf(v.z, v.w));
  #pragma unroll
  for (int off = 16; off > 0; off >>= 1) m = fmaxf(m, __shfl_xor(m, off, 32));
  if (lane == 0) red[wv] = m;
  __syncthreads();
  m = red[0];
  #pragma unroll
  for (int i = 1; i < 8; ++i) m = fmaxf(m, red[i]);
  __syncthreads();

  float4 e;
  e.x = __expf(v.x - m); e.y = __expf(v.y - m);
  e.z = __expf(v.z - m); e.w = __expf(v.w - m);
  float ssum = e.x + e.y + e.z + e.w;
  #pragma unroll
  for (int off = 16; off > 0; off >>= 1) ssum += __shfl_xor(ssum, off, 32);
  if (lane == 0) red[wv] = ssum;
  __syncthreads();
  float tot = 0.0f;
  #pragma unroll
  for (int i = 0; i < 8; ++i) tot += red[i];
  const float inv = 1.0f / tot;
  e.x *= inv; e.y *= inv; e.z *= inv; e.w *= inv;
  ((float4*)wr)[tid] = e;
}

// ---------------------------------------------------------------------------
// Kernel 4: O = w @ V per (b,h) -> AO in [B, N, H*Dh] row-major.
// Each wave: 16 rows x full 64-wide head.  grid = (1024/128, 96)
// ---------------------------------------------------------------------------
__global__ __launch_bounds__(256) void av_kernel(
    const float* __restrict__ Wm, const float* __restrict__ V,
    float* __restrict__ AO)
{
  const int bh = blockIdx.y;
  const int bb = bh / H_CNT;
  const int h  = bh - bb * H_CNT;
  const float* w = Wm + (size_t)bh * N_SEQ * N_SEQ;
  const float* v = V  + (size_t)bh * N_SEQ * D_HEAD;

  const int lane = threadIdx.x & 31;
  const int wave = threadIdx.x >> 5;
  const int l    = lane & 15;
  const int hf   = lane >> 4;
  const int row0 = blockIdx.x * 128 + wave * 16;

  v8f acc[4] = {{}, {}, {}, {}};
  const float* arow = w + (size_t)(row0 + l) * N_SEQ;
  for (int k = 0; k < N_SEQ; k += 4) {
    const int kk = k + 2 * hf;
    const v2f a = *(const v2f*)(arow + kk);
    const float* b0 = v + (size_t)kk * D_HEAD + l;
    const float* b1 = b0 + D_HEAD;
    #pragma unroll
    for (int t = 0; t < 4; ++t) {
      v2f b; b.x = b0[16 * t]; b.y = b1[16 * t];
      acc[t] = wmma4(a, b, acc[t]);
    }
  }
  #pragma unroll
  for (int t = 0; t < 4; ++t) {
    #pragma unroll
    for (int j = 0; j < 8; ++j) {
      const int n = row0 + j + 8 * hf;
      AO[(size_t)(bb * N_SEQ + n) * D_MOD + h * D_HEAD + 16 * t + l] = acc[t][j];
    }
  }
}

// ---------------------------------------------------------------------------
// Kernel 5: out = AO[8192,768] @ Wproj[768,768] + bproj.
// grid = (768/128, 8192/64)
// ---------------------------------------------------------------------------
__global__ __launch_bounds__(256) void proj_kernel(
    const float* __restrict__ A, const float* __restrict__ W,
    const float* __restrict__ bias, float* __restrict__ O)
{
  const int lane = threadIdx.x & 31;
  const int wave = threadIdx.x >> 5;
  const int l    = lane & 15;
  const int hf   = lane >> 4;
  const int row0 = blockIdx.y * 64 + (wave >> 1) * 16;
  const int col0 = blockIdx.x * 128 + (wave & 1) * 64;

  v8f acc[4] = {{}, {}, {}, {}};
  const float* arow = A + (size_t)(row0 + l) * D_MOD;
  for (int k = 0; k < D_MOD; k += 4) {
    const int kk = k + 2 * hf;
    const v2f a = *(const v2f*)(arow + kk);
    const float* b0 = W + (size_t)kk * D_MOD + col0 + l;
    const float* b1 = b0 + D_MOD;
    #pragma unroll
    for (int t = 0; t < 4; ++t) {
      v2f b; b.x = b0[16 * t]; b.y = b1[16 * t];
      acc[t] = wmma4(a, b, acc[t]);
    }
  }
  #pragma unroll
  for (int t = 0; t < 4; ++t) {
    const float bv = bias[col0 + 16 * t + l];
    #pragma unroll
    for (int j = 0; j < 8; ++j) {
      const int r = row0 + j + 8 * hf;
      O[(size_t)r * D_MOD + col0 + 16 * t + l] = acc[t][j] + bv;
    }
  }
}

// ---------------------------------------------------------------------------
extern "C" void kernel_launch(void* const* d_in, const int* in_sizes, int n_in,
                              void* d_out, int out_size, void* d_ws,
                              size_t ws_size, hipStream_t stream)
{
  (void)in_sizes; (void)n_in; (void)out_size; (void)ws_size;
  const float* x     = (const float*)d_in[0];
  const float* Wqkv  = (const float*)d_in[1];
  const float* Wproj = (const float*)d_in[2];
  const float* bproj = (const float*)d_in[3];
  const float* alpha = (const float*)d_in[4];

  const size_t CH = (size_t)B_SZ * H_CNT * N_SEQ * D_HEAD; // 6,291,456 floats
  float* ws = (float*)d_ws;
  float* Qb = ws;
  float* Kb = ws + CH;
  float* Vb = ws + 2 * CH;
  float* AO = ws + 3 * CH;
  unsigned* amax = (unsigned*)(ws + 4 * CH);

  float* out = (float*)d_out;                           // [8,1024,768]
  float* wat = out + (size_t)B_SZ * N_SEQ * D_MOD;      // [8,12,1024,1024]

  qkv_gemm_kernel<<<dim3(N_QKV / 128, (B_SZ * N_SEQ) / 64), 256, 0, stream>>>(
      x, Wqkv, Qb, Kb, Vb);
  init_absmax_kernel<<<1, 128, 0, stream>>>(amax);
  scores_kernel<<<dim3(N_SEQ / 128, N_SEQ / 64, B_SZ * H_CNT), 256, 0, stream>>>(
      Qb, Kb, wat, amax);
  softmax_kernel<<<dim3(B_SZ * H_CNT * N_SEQ), 256, 0, stream>>>(wat, amax, alpha);
  av_kernel<<<dim3(N_SEQ / 128, B_SZ * H_CNT), 256, 0, stream>>>(wat, Vb, AO);
  proj_kernel<<<dim3(D_MOD / 128, (B_SZ * N_SEQ) / 64), 256, 0, stream>>>(
      AO, Wproj, bproj, out);
}